// PaperModel_30889404793005
// MI455X (gfx1250) — compile-verified
//
#include <hip/hip_runtime.h>
#include <math.h>

// Problem constants from setup_inputs(): n=64, k=5, q=128, d=256
#define NCLS   64
#define KSUP   5
#define QPER   128
#define DIM    256
#define NQ     (NCLS * QPER)      // 8192
#define ROWLEN (KSUP + QPER)      // 133
#define KNEIGH 10

typedef float v2f __attribute__((ext_vector_type(2)));
typedef float v8f __attribute__((ext_vector_type(8)));

// ---------------- block reductions (blockDim.x == 256) ----------------
__device__ __forceinline__ float block_sum(float v, float* red) {
    int t = threadIdx.x;
    red[t] = v; __syncthreads();
    for (int s = 128; s > 0; s >>= 1) {
        if (t < s) red[t] += red[t + s];
        __syncthreads();
    }
    float r = red[0]; __syncthreads();
    return r;
}
__device__ __forceinline__ float block_max(float v, float* red) {
    int t = threadIdx.x;
    red[t] = v; __syncthreads();
    for (int s = 128; s > 0; s >>= 1) {
        if (t < s) red[t] = fmaxf(red[t], red[t + s]);
        __syncthreads();
    }
    float r = red[0]; __syncthreads();
    return r;
}

// ---------------- prototype mean + normalize ----------------
__global__ __launch_bounds__(256) void proto_kernel(const float* __restrict__ x,
                                                    float* __restrict__ proto,
                                                    float* __restrict__ pn) {
    __shared__ float red[256];
    int c = blockIdx.x, t = threadIdx.x;
    float s = 0.f;
    for (int j = 0; j < KSUP; ++j)
        s += x[(size_t)(c * ROWLEN + j) * DIM + t];
    float p = s * (1.0f / KSUP);
    proto[c * DIM + t] = p;
    float nrm = sqrtf(block_sum(p * p, red));
    pn[c * DIM + t] = p / nrm;
}

// ---------------- copy + normalize query rows ----------------
__global__ __launch_bounds__(256) void query_norm_kernel(const float* __restrict__ x,
                                                         float* __restrict__ query,
                                                         float* __restrict__ qn) {
    __shared__ float red[256];
    int g = blockIdx.x, t = threadIdx.x;
    int cls = g >> 7, j = g & (QPER - 1);
    float v = x[(size_t)(cls * ROWLEN + KSUP + j) * DIM + t];
    query[(size_t)g * DIM + t] = v;
    float nrm = sqrtf(block_sum(v * v, red));
    qn[(size_t)g * DIM + t] = v / nrm;
}

// ---------------- generic C = scale * A(MxK) @ B(NxK)^T via f32 WMMA ----------------
// grid = (N/16, M/16), block = 32 (one wave per 16x16 tile)
__global__ __launch_bounds__(32) void gemm_nt_wmma(const float* __restrict__ A,
                                                   const float* __restrict__ B,
                                                   float* __restrict__ C,
                                                   int N, int K,
                                                   const float* __restrict__ scale_ptr) {
    int lane = threadIdx.x;
    int r16  = lane & 15;
    int koff = (lane < 16) ? 0 : 2;
    const float* Arow = A + (size_t)(blockIdx.y * 16 + r16) * K;
    const float* Brow = B + (size_t)(blockIdx.x * 16 + r16) * K;
    v8f c = {};
    for (int kk = 0; kk < K; kk += 4) {
        v2f a, b;
        a.x = Arow[kk + koff];     a.y = Arow[kk + koff + 1];
        b.x = Brow[kk + koff];     b.y = Brow[kk + koff + 1];
        c = __builtin_amdgcn_wmma_f32_16x16x4_f32(false, a, false, b,
                                                  (short)0, c, false, false);
    }
    float s = scale_ptr ? scale_ptr[0] : 1.0f;
    int mbase = blockIdx.y * 16 + ((lane < 16) ? 0 : 8);
    int nidx  = blockIdx.x * 16 + r16;
#pragma unroll
    for (int v = 0; v < 8; ++v)
        C[(size_t)(mbase + v) * N + nidx] = s * c[v];
}

// ---------------- fused query_sim (WMMA) + streaming per-row top-10 ----------------
// grid = NQ/16 blocks, block = 256 (8 waves). Each block owns 16 rows; the A
// fragments for the whole K=256 loop are hoisted into 128 VGPRs/lane and reused
// across all 64 column chunks (only B is streamed from L2). Each 16x128 sim
// chunk goes through LDS; 16 threads per row keep private top-10 lists over
// 8-candidate strided slices (all 256 lanes busy), merged exactly at the end.
__global__ __launch_bounds__(256) void topk_kernel(const float* __restrict__ qn,
                                                   int* __restrict__ kind,
                                                   float* __restrict__ kval) {
    __shared__ float simbuf[16][132];          // padded stride: avoid bank conflicts
    __shared__ float ptv[256][KNEIGH];         // per-thread partial top-10 values
    __shared__ int   pti[256][KNEIGH];         // per-thread partial top-10 indices
    int tid  = threadIdx.x;
    int wave = tid >> 5;
    int lane = tid & 31;
    int r16  = lane & 15;
    int koff = (lane < 16) ? 0 : 2;
    int r0   = blockIdx.x * 16;
    int srow = tid >> 4;                       // row this thread helps scan
    int ssub = tid & 15;                       // sub-slice within the row

#pragma unroll
    for (int j = 0; j < KNEIGH; ++j) { ptv[tid][j] = -INFINITY; pti[tid][j] = 0; }

    // Hoist A fragments for the full K loop: 64 x v2f = 128 VGPRs per lane.
    const float* Arow = qn + (size_t)(r0 + r16) * DIM;
    v2f areg[DIM / 4];
#pragma unroll
    for (int kk = 0; kk < DIM / 4; ++kk) {
        areg[kk].x = Arow[kk * 4 + koff];
        areg[kk].y = Arow[kk * 4 + koff + 1];
    }

    for (int cb = 0; cb < NQ / 128; ++cb) {
        const float* Brow = qn + (size_t)(cb * 128 + wave * 16 + r16) * DIM;
        v8f c = {};
#pragma unroll
        for (int kk = 0; kk < DIM / 4; ++kk) {
            v2f b;
            b.x = Brow[kk * 4 + koff];
            b.y = Brow[kk * 4 + koff + 1];
            c = __builtin_amdgcn_wmma_f32_16x16x4_f32(false, areg[kk], false, b,
                                                      (short)0, c, false, false);
        }
        int mbase = (lane < 16) ? 0 : 8;
#pragma unroll
        for (int v = 0; v < 8; ++v)
            simbuf[mbase + v][wave * 16 + r16] = c[v];
        __syncthreads();

        // each thread scans cols cc = ssub + 16*s (increasing global index ->
        // stable lowest-index-first tie handling within each private list)
        for (int s = 0; s < 8; ++s) {
            int cc = ssub + 16 * s;
            float val = simbuf[srow][cc];
            if (val > ptv[tid][KNEIGH - 1]) {
                int col = cb * 128 + cc;
                int p = KNEIGH - 1;
                while (p > 0 && ptv[tid][p - 1] < val) {
                    ptv[tid][p] = ptv[tid][p - 1];
                    pti[tid][p] = pti[tid][p - 1];
                    --p;
                }
                ptv[tid][p] = val;
                pti[tid][p] = col;
            }
        }
        __syncthreads();
    }

    // 16-way merge per row (value desc, index asc == jax.lax.top_k tie-break)
    if (tid < 16) {
        int row = tid;
        int hp[16];
#pragma unroll
        for (int u = 0; u < 16; ++u) hp[u] = 0;
        for (int outj = 0; outj < KNEIGH; ++outj) {
            float bv = -INFINITY; int bi = 0x7fffffff; int bu = 0;
            for (int u = 0; u < 16; ++u) {
                if (hp[u] < KNEIGH) {
                    float v  = ptv[row * 16 + u][hp[u]];
                    int   ix = pti[row * 16 + u][hp[u]];
                    if (v > bv || (v == bv && ix < bi)) { bv = v; bi = ix; bu = u; }
                }
            }
            hp[bu]++;
            kind[(size_t)(r0 + row) * KNEIGH + outj] = bi;
            kval[(size_t)(r0 + row) * KNEIGH + outj] = bv;
        }
    }
}

// ---------------- per-query argmax over 64 classes ----------------
__global__ __launch_bounds__(256) void argmax_kernel(const float* __restrict__ presim,
                                                     int* __restrict__ label) {
    int i = blockIdx.x * blockDim.x + threadIdx.x;
    if (i >= NQ) return;
    const float* row = presim + (size_t)i * NCLS;
    float best = row[0]; int bi = 0;
    for (int c = 1; c < NCLS; ++c) {
        float v = row[c];
        if (v > best) { best = v; bi = c; }
    }
    label[i] = bi;
}

// ---------------- per-class masked softmax -> adapted & normalized proto ----------------
__global__ __launch_bounds__(256) void class_stats_kernel(const float* __restrict__ pn,
                                                          const float* __restrict__ proto,
                                                          const float* __restrict__ presim,
                                                          const int* __restrict__ label,
                                                          const float* __restrict__ query,
                                                          float* __restrict__ apn) {
    __shared__ float red[256];
    int c = blockIdx.x, t = threadIdx.x;
    float pv = pn[c * DIM + t];
    float self_sim = block_sum(pv * pv, red);

    float mx = -INFINITY;
    for (int i = t; i < NQ; i += 256)
        if (label[i] == c) mx = fmaxf(mx, presim[(size_t)i * NCLS + c]);
    float m = fmaxf(block_max(mx, red), self_sim);

    float es  = __expf(self_sim - m);
    float acc = es * proto[c * DIM + t];
    float Z   = es;
    for (int i = 0; i < NQ; ++i) {
        if (label[i] == c) {
            float e = __expf(presim[(size_t)i * NCLS + c] - m);
            Z   += e;
            acc += e * query[(size_t)i * DIM + t];
        }
    }
    float ap  = acc / Z;
    float nrm = sqrtf(block_sum(ap * ap, red));
    apn[c * DIM + t] = ap / nrm;
}

// ---------------- mutual top-k softmax -> adapted & normalized query ----------------
__global__ __launch_bounds__(256) void mutual_adapt_kernel(const int* __restrict__ kind,
                                                           const float* __restrict__ kval,
                                                           const float* __restrict__ query,
                                                           float* __restrict__ aqn) {
    __shared__ float red[256];
    __shared__ int   nbi[KNEIGH];
    __shared__ float nbw[KNEIGH];
    int i = blockIdx.x, t = threadIdx.x;

    if (t < KNEIGH) {
        int j = kind[(size_t)i * KNEIGH + t];
        bool mut = false;
        for (int s = 0; s < KNEIGH; ++s)
            mut |= (kind[(size_t)j * KNEIGH + s] == i);
        nbi[t] = j;
        nbw[t] = mut ? kval[(size_t)i * KNEIGH + t] : -INFINITY;
    }
    __syncthreads();
    if (t == 0) {
        float m = -INFINITY;
        for (int s = 0; s < KNEIGH; ++s) m = fmaxf(m, nbw[s]);
        float e[KNEIGH]; float Z = 0.f;
        for (int s = 0; s < KNEIGH; ++s) {
            e[s] = (nbw[s] == -INFINITY) ? 0.f : __expf(nbw[s] - m);
            Z += e[s];
        }
        for (int s = 0; s < KNEIGH; ++s) nbw[s] = e[s] / Z;
    }
    __syncthreads();

    float aq = 0.f;
    for (int s = 0; s < KNEIGH; ++s)
        aq += nbw[s] * query[(size_t)nbi[s] * DIM + t];
    float nrm = sqrtf(block_sum(aq * aq, red));
    aqn[(size_t)i * DIM + t] = aq / nrm;
}

// ---------------- host-side launch ----------------
extern "C" void kernel_launch(void* const* d_in, const int* in_sizes, int n_in,
                              void* d_out, int out_size, void* d_ws, size_t ws_size,
                              hipStream_t stream) {
    const float* x   = (const float*)d_in[0];
    const float* tao = (const float*)d_in[1];   // device scalar
    (void)in_sizes; (void)n_in; (void)out_size; (void)ws_size;

    float* ws     = (float*)d_ws;
    float* query  = ws;                         // NQ*DIM
    float* qn     = query  + (size_t)NQ * DIM;  // NQ*DIM
    float* proto  = qn     + (size_t)NQ * DIM;  // NCLS*DIM
    float* pn     = proto  + (size_t)NCLS * DIM;
    float* presim = pn     + (size_t)NCLS * DIM;            // NQ*NCLS
    int*   label  = (int*)(presim + (size_t)NQ * NCLS);     // NQ
    int*   kind   = label  + NQ;                            // NQ*KNEIGH
    float* kval   = (float*)(kind + (size_t)NQ * KNEIGH);   // NQ*KNEIGH
    float* aqn    = kval   + (size_t)NQ * KNEIGH;           // NQ*DIM
    float* apn    = aqn    + (size_t)NQ * DIM;              // NCLS*DIM

    proto_kernel<<<NCLS, 256, 0, stream>>>(x, proto, pn);
    query_norm_kernel<<<NQ, 256, 0, stream>>>(x, query, qn);

    // pre_sim = qn @ pn^T  (8192x64)
    gemm_nt_wmma<<<dim3(NCLS / 16, NQ / 16), 32, 0, stream>>>(qn, pn, presim,
                                                              NCLS, DIM, nullptr);
    argmax_kernel<<<NQ / 256, 256, 0, stream>>>(presim, label);
    class_stats_kernel<<<NCLS, 256, 0, stream>>>(pn, proto, presim, label, query, apn);

    // fused query_sim + top-10 (never materializes 8192x8192)
    topk_kernel<<<NQ / 16, 256, 0, stream>>>(qn, kind, kval);
    mutual_adapt_kernel<<<NQ, 256, 0, stream>>>(kind, kval, query, aqn);

    // out = tao * (aqn @ apn^T)  (8192x64)
    gemm_nt_wmma<<<dim3(NCLS / 16, NQ / 16), 32, 0, stream>>>(aqn, apn, (float*)d_out,
                                                              NCLS, DIM, tao);
}